// NNConv1Layer_61632780698130
// MI455X (gfx1250) — compile-verified
//
#include <hip/hip_runtime.h>
#include <hip/hip_bf16.h>
#include <math.h>

typedef __attribute__((ext_vector_type(2))) float v2f;
typedef __attribute__((ext_vector_type(8))) float v8f;

#define Z_STRIDE 36  // 32 z cols + 1 bias col + pad (36*4 = 144 B, 16B-aligned rows)
#define MT 4         // M-tiles (of 16 rows) per wave -> 64 rows/wave

// ---------------------------------------------------------------------------
// z[N,36] = x[N,F] @ nn_w[F,32]  (cols 0..31),  col 32 = x @ nn_b
// One wave computes 4 16-row tiles via V_WMMA_F32_16X16X4_F32; B fragments are
// loaded once per k-step and reused across the 4 tiles (12 WMMAs per k-step).
// A fragments are loaded as a batch so the 4 loads clause together and drain
// with staggered s_wait_loadcnt while earlier WMMAs execute.
// ---------------------------------------------------------------------------
__global__ void nnw_gemm_wmma_kernel(const float* __restrict__ x,
                                     const float* __restrict__ nw,   // [F,32] row-major
                                     const float* __restrict__ nb,   // [F]
                                     float* __restrict__ z,
                                     int N, int F) {
    const int wave = (blockIdx.x * blockDim.x + threadIdx.x) >> 5;
    const int lane = threadIdx.x & 31;
    const int row0 = wave * (16 * MT);
    if (row0 >= N) return;                   // uniform per wave

    const int khalf = (lane >> 4) * 2;       // lanes 0-15: K+0/K+1, lanes 16-31: K+2/K+3
    const int col   = lane & 15;

    // per-tile A row (clamped so tail-wave loads stay in bounds; EXEC stays full)
    int mrow[MT];
#pragma unroll
    for (int t = 0; t < MT; ++t) {
        int r = row0 + t * 16 + (lane & 15);
        mrow[t] = (r < N) ? r : (N - 1);
    }

    v8f c0[MT], c1[MT], cb[MT];
#pragma unroll
    for (int t = 0; t < MT; ++t) { c0[t] = (v8f){}; c1[t] = (v8f){}; cb[t] = (v8f){}; }

    for (int k = 0; k < F; k += 4) {
        const int ka = k + khalf;
        // B fragments: 4x16 tiles of nn_w (cols 0-15 and 16-31); bias tile col 0
        v2f b0, b1, bb;
        b0.x = nw[(size_t)ka * 32 + col];
        b0.y = nw[(size_t)(ka + 1) * 32 + col];
        b1.x = nw[(size_t)ka * 32 + col + 16];
        b1.y = nw[(size_t)(ka + 1) * 32 + col + 16];
        const float nb0 = nb[ka];            // unconditional broadcast loads
        const float nb1 = nb[ka + 1];
        bb.x = (col == 0) ? nb0 : 0.0f;      // v_cndmask, no predicated load
        bb.y = (col == 0) ? nb1 : 0.0f;

        // batch the 4 A-fragment loads (one clause), then issue the 12 WMMAs
        v2f a[MT];
#pragma unroll
        for (int t = 0; t < MT; ++t) {
            const float2 av = *(const float2*)(x + (size_t)mrow[t] * F + ka);
            a[t].x = av.x; a[t].y = av.y;
        }
#pragma unroll
        for (int t = 0; t < MT; ++t) {
            c0[t] = __builtin_amdgcn_wmma_f32_16x16x4_f32(false, a[t], false, b0, (short)0, c0[t], false, false);
            c1[t] = __builtin_amdgcn_wmma_f32_16x16x4_f32(false, a[t], false, b1, (short)0, c1[t], false, false);
            cb[t] = __builtin_amdgcn_wmma_f32_16x16x4_f32(false, a[t], false, bb, (short)0, cb[t], false, false);
        }
    }

    // C/D layout: VGPR i -> (M=i, N=lane) lanes 0-15; (M=8+i, N=lane-16) lanes 16-31
    const int cr = (lane < 16) ? 0 : 8;
#pragma unroll
    for (int t = 0; t < MT; ++t) {
        const int tr = row0 + t * 16;
        if (tr < N) {                        // wave-uniform guard (scalar branch)
#pragma unroll
            for (int i = 0; i < 8; ++i) {
                const size_t r = (size_t)(tr + cr + i) * Z_STRIDE;
                z[r + col]      = c0[t][i];
                z[r + 16 + col] = c1[t][i];
                if (col == 0) z[r + 32] = cb[t][i];
            }
        }
    }
}

// ---------------------------------------------------------------------------
// fe==1 edge types: z[row,0] = x_row . nn_w[:,0],  z[row,32] = x_row . nn_b
// ---------------------------------------------------------------------------
__global__ void node_dot2_kernel(const float* __restrict__ x,
                                 const float* __restrict__ nw,   // [F,1] == contiguous F
                                 const float* __restrict__ nb,   // [F]
                                 float* __restrict__ z,
                                 int N, int F) {
    const int wave = (blockIdx.x * blockDim.x + threadIdx.x) >> 5;
    const int lane = threadIdx.x & 31;
    if (wave >= N) return;
    float s0 = 0.f, s1 = 0.f;
    for (int k = lane; k < F; k += 32) {
        const float xv = x[(size_t)wave * F + k];
        s0 += xv * nw[k];
        s1 += xv * nb[k];
    }
#pragma unroll
    for (int off = 16; off; off >>= 1) {
        s0 += __shfl_xor(s0, off);
        s1 += __shfl_xor(s1, off);
    }
    if (lane == 0) {
        z[(size_t)wave * Z_STRIDE]      = s0;
        z[(size_t)wave * Z_STRIDE + 32] = s1;
    }
}

// ---------------------------------------------------------------------------
// fe==32 edge messages: one wave per edge (fully coalesced 128B ea + z loads).
// msg = ea[e,:32] . z[src,:32] + z[src,32];  atomicAdd into accum[dst]
// ---------------------------------------------------------------------------
__global__ void edge_msg32_kernel(const int* __restrict__ src,
                                  const int* __restrict__ dst,
                                  const float* __restrict__ ea,
                                  const float* __restrict__ z,
                                  float* __restrict__ accum, int E) {
    const int e    = (blockIdx.x * blockDim.x + threadIdx.x) >> 5;
    const int lane = threadIdx.x & 31;
    if (e >= E) return;
    const int s = src[e];
    const int d = dst[e];                         // unconditional (broadcast)
    // prefetch the edge-attr stream a few edges ahead (clamped, in-bounds)
    const int pe = (e + 64 < E) ? (e + 64) : (E - 1);
    __builtin_prefetch(ea + (size_t)pe * 32 + lane, 0, 1);

    float m = ea[(size_t)e * 32 + lane] * z[(size_t)s * Z_STRIDE + lane];
    const float bias = z[(size_t)s * Z_STRIDE + 32];  // broadcast, unconditional
#pragma unroll
    for (int off = 16; off; off >>= 1) m += __shfl_xor(m, off);
    if (lane == 0) {
        atomicAdd(&accum[d], m + bias);
    }
}

// fe==1 edge messages: one thread per edge.
__global__ void edge_msg1_kernel(const int* __restrict__ src,
                                 const int* __restrict__ dst,
                                 const float* __restrict__ ea,
                                 const float* __restrict__ z,
                                 float* __restrict__ accum, int E) {
    const int e = blockIdx.x * blockDim.x + threadIdx.x;
    if (e >= E) return;
    const int s = src[e];
    const float m = ea[e] * z[(size_t)s * Z_STRIDE] + z[(size_t)s * Z_STRIDE + 32];
    atomicAdd(&accum[dst[e]], m);
}

// ---------------------------------------------------------------------------
// Combined root weights: W[0..127] = sum of 7 lin_w; W[128] = sum of 7 biases
// ---------------------------------------------------------------------------
__global__ void combine_root_kernel(float* __restrict__ W,
                                    const float* w0, const float* w1, const float* w2,
                                    const float* w3, const float* w4, const float* w5,
                                    const float* w6,
                                    const float* b0, const float* b1, const float* b2,
                                    const float* b3, const float* b4, const float* b5,
                                    const float* b6, int F) {
    const int t = threadIdx.x;
    if (t < F) W[t] = w0[t] + w1[t] + w2[t] + w3[t] + w4[t] + w5[t] + w6[t];
    if (t == 0) W[F] = b0[0] + b1[0] + b2[0] + b3[0] + b4[0] + b5[0] + b6[0];
}

// ---------------------------------------------------------------------------
// out[i] = sigmoid(accum[i] + x[i,:] . W[0..F) + W[F])   (wave per row)
// ---------------------------------------------------------------------------
__global__ void finalize_kernel(const float* __restrict__ x,
                                const float* __restrict__ accum,
                                const float* __restrict__ W,
                                float* __restrict__ out, int N, int F) {
    const int row  = (blockIdx.x * blockDim.x + threadIdx.x) >> 5;
    const int lane = threadIdx.x & 31;
    if (row >= N) return;
    float s = 0.f;
    for (int k = lane; k < F; k += 32) s += x[(size_t)row * F + k] * W[k];
    const float acc = accum[row];                 // unconditional broadcast
    const float bw  = W[F];
#pragma unroll
    for (int off = 16; off; off >>= 1) s += __shfl_xor(s, off);
    if (lane == 0) {
        const float v = acc + s + bw;
        out[row] = 1.0f / (1.0f + expf(-v));
    }
}

// ---------------------------------------------------------------------------
// Host launch
// ---------------------------------------------------------------------------
struct ETInfo { int s, d, fe, E; };
static const ETInfo g_et[14] = {
    {0,0,32,150000},{1,0,32,150000},{2,0,32,150000},
    {0,1,32,150000},{1,1,32,150000},{2,1,32,150000},
    {0,1, 1,100000},
    {0,0,32,150000},{1,0,32,150000},{2,0,32,150000},
    {0,1,32,150000},{1,1,32,150000},{2,1,32,150000},
    {1,0, 1,100000},
};
static const int g_NN[3] = {100000, 20000, 10000};
static const int g_NF[3] = {128, 128, 64};

extern "C" void kernel_launch(void* const* d_in, const int* in_sizes, int n_in,
                              void* d_out, int out_size, void* d_ws, size_t ws_size,
                              hipStream_t stream) {
    (void)in_sizes; (void)n_in; (void)out_size; (void)ws_size;
    const float* x[3] = {(const float*)d_in[0], (const float*)d_in[1], (const float*)d_in[2]};

    // d_in layout (dict order): x_ind,x_org,x_ext | 14x(src,dst) | 14x ea | 14x(nn_w,nn_b,lin_w,bias)
    float* ws = (float*)d_ws;
    float* accum_ind = ws;              // 100000
    float* accum_org = ws + 100000;     // 20000
    float* W_ind     = ws + 120000;     // 129 (pad to 120132)
    float* W_org     = ws + 120132;     // 129 (pad to 120264)
    float* z         = ws + 120264;     // 100000 * 36 floats (reused per edge type)

    // zero accumulators (ind+org)
    hipMemsetAsync(ws, 0, (size_t)120000 * sizeof(float), stream);

    // combined root terms
    const int indEt[7] = {0,1,2,7,8,9,13};
    const int orgEt[7] = {3,4,5,6,10,11,12};
#define LW(i) ((const float*)d_in[47 + 4*(i)])
#define BI(i) ((const float*)d_in[48 + 4*(i)])
    combine_root_kernel<<<1, 128, 0, stream>>>(W_ind,
        LW(indEt[0]), LW(indEt[1]), LW(indEt[2]), LW(indEt[3]), LW(indEt[4]), LW(indEt[5]), LW(indEt[6]),
        BI(indEt[0]), BI(indEt[1]), BI(indEt[2]), BI(indEt[3]), BI(indEt[4]), BI(indEt[5]), BI(indEt[6]), 128);
    combine_root_kernel<<<1, 128, 0, stream>>>(W_org,
        LW(orgEt[0]), LW(orgEt[1]), LW(orgEt[2]), LW(orgEt[3]), LW(orgEt[4]), LW(orgEt[5]), LW(orgEt[6]),
        BI(orgEt[0]), BI(orgEt[1]), BI(orgEt[2]), BI(orgEt[3]), BI(orgEt[4]), BI(orgEt[5]), BI(orgEt[6]), 128);
#undef LW
#undef BI

    for (int i = 0; i < 14; ++i) {
        const ETInfo& et = g_et[i];
        const int*   src = (const int*)d_in[3 + 2*i];
        const int*   dst = (const int*)d_in[4 + 2*i];
        const float* ea  = (const float*)d_in[31 + i];
        const float* nw  = (const float*)d_in[45 + 4*i];
        const float* nb  = (const float*)d_in[46 + 4*i];
        const int Ns = g_NN[et.s], Fs = g_NF[et.s];
        float* accum = (et.d == 0) ? accum_ind : accum_org;

        if (et.fe == 32) {
            const int waves = (Ns + 16 * MT - 1) / (16 * MT);
            const int thr   = waves * 32;
            nnw_gemm_wmma_kernel<<<(thr + 255) / 256, 256, 0, stream>>>(x[et.s], nw, nb, z, Ns, Fs);
            const long long ethr = (long long)et.E * 32;
            edge_msg32_kernel<<<(unsigned)((ethr + 255) / 256), 256, 0, stream>>>(src, dst, ea, z, accum, et.E);
        } else {
            const int thr = Ns * 32;
            node_dot2_kernel<<<(thr + 255) / 256, 256, 0, stream>>>(x[et.s], nw, nb, z, Ns, Fs);
            edge_msg1_kernel<<<(et.E + 255) / 256, 256, 0, stream>>>(src, dst, ea, z, accum, et.E);
        }
    }

    float* out = (float*)d_out;
    finalize_kernel<<<((100000 * 32) + 255) / 256, 256, 0, stream>>>(x[0], accum_ind, W_ind, out, 100000, 128);
    finalize_kernel<<<((20000 * 32) + 255) / 256, 256, 0, stream>>>(x[1], accum_org, W_org, out + 100000, 20000, 128);
}